// FixedPtSQP_70489003262098
// MI455X (gfx1250) — compile-verified
//
#include <hip/hip_runtime.h>
#include <hip/hip_bf16.h>

// FixedPtSQP on MI455X (gfx1250, wave32).
//
// Structure exploited: Q=I, G=[I;-I], A=1^T  =>  the per-iteration 65x65 KKT
// solve collapses to a diagonal solve + scalar Schur complement. One wave32
// per batch element; all IPM state lives in VGPRs.
//
// Cross-lane ops:
//   * wave-wide f32 sums run on the matrix pipe (V_WMMA_F32_16X16X4_F32 with
//     B = ones => D[m,n] = u_m + u_{m+16}), finished with one permlanex16.
//   * wave-wide min is a pure-VALU butterfly (DPP quad_perm / half-mirror /
//     mirror + permlanex16) — no LDS traffic in the hot loop.
// Divisions: v_rcp_f32 (+1 Newton step where the result feeds the solve),
// eliminating the divergent IEEE-div slow paths (~23 divides/iteration).

typedef float v8f __attribute__((ext_vector_type(8)));
typedef float v2f __attribute__((ext_vector_type(2)));

#define N_VARS   64
#define SQP_IT   10
#define IPM_IT   25
#define ALPHA_   0.5f
#define SIGMA_   0.1f
#define M_INEQ   128

// ---- fast reciprocals ------------------------------------------------------

// raw v_rcp_f32 (~1 ulp) — used only in the step-length ratio test
__device__ __forceinline__ float rcp_fast(float x) {
    return __builtin_amdgcn_rcpf(x);
}
// v_rcp_f32 + 1 Newton step (~0.5 ulp) — used where the value feeds the solve
__device__ __forceinline__ float rcp_acc(float x) {
    float r = __builtin_amdgcn_rcpf(x);
    return fmaf(fmaf(-x, r, 1.0f), r, r);
}

// ---- VALU cross-lane primitives (no LDS) -----------------------------------

// DPP16 lane permute within 16-lane rows. ctrl: 0x00-0xFF quad_perm,
// 0x141 row_half_mirror (l^7), 0x140 row_mirror (l^15).
template <int CTRL>
__device__ __forceinline__ float dpp_mov(float x) {
    int i = __float_as_int(x);
    int r = __builtin_amdgcn_update_dpp(i, i, CTRL, 0xF, 0xF, true);
    return __int_as_float(r);
}

// Swap the two 16-lane halves (lane l <-> lane l^16) on the VALU pipe.
__device__ __forceinline__ float permx16(float x) {
    int i = __float_as_int(x);
    int r = __builtin_amdgcn_permlanex16(i, i, 0x76543210, (int)0xFEDCBA98,
                                         false, false);
    return __int_as_float(r);
}

// Sum of `u` across all 32 lanes, result broadcast to every lane.
// A(16x4) = {u_m in even-K slots, 0 in odd-K}; B = all-ones. Then
// D[m,n] = u_m + u_{m+16} regardless of the B register layout. Summing D's 8
// VGPRs covers rows {r, r+8} per lane half; permlanex16 folds the halves.
__device__ __forceinline__ float wave_sum(float u) {
    v2f a; a.x = u;    a.y = 0.0f;
    v2f b; b.x = 1.0f; b.y = 1.0f;
    v8f cz = {};
    v8f d = __builtin_amdgcn_wmma_f32_16x16x4_f32(
        /*neg_a=*/false, a, /*neg_b=*/false, b,
        /*c_mod=*/(short)0, cz, /*reuse_a=*/false, /*reuse_b=*/false);
    float t = ((d[0] + d[1]) + (d[2] + d[3])) + ((d[4] + d[5]) + (d[6] + d[7]));
    t += permx16(t);
    return t;   // identical in all lanes
}

// Min across 32 lanes, broadcast. Butterfly stages:
//   quad_perm xor1, quad_perm xor2  -> min over groups of 4
//   row_half_mirror (l^7)           -> min over groups of 8  (bit2 flips)
//   row_mirror     (l^15)           -> min over groups of 16 (bit3 flips)
//   permlanex16    (l^16)           -> min over all 32
__device__ __forceinline__ float wave_min(float u) {
    u = fminf(u, dpp_mov<0xB1>(u));    // quad_perm(1,0,3,2)  : xor 1
    u = fminf(u, dpp_mov<0x4E>(u));    // quad_perm(2,3,0,1)  : xor 2
    u = fminf(u, dpp_mov<0x141>(u));   // row_half_mirror     : xor 7
    u = fminf(u, dpp_mov<0x140>(u));   // row_mirror          : xor 15
    u = fminf(u, permx16(u));          // cross-half          : xor 16
    return u;
}

// fraction-to-boundary ratio: dv<0 ? -v/dv : +inf   ( -v/dv = v * rcp(-dv) )
__device__ __forceinline__ float ftb(float v, float dv) {
    return (dv < 0.0f) ? (v * rcp_fast(-dv)) : __builtin_inff();
}

// ---- kernel ----------------------------------------------------------------

__global__ __launch_bounds__(256)
void FixedPtSQP_kernel(const float* __restrict__ c_in,
                       float* __restrict__ out,
                       int batch) {
    const int lane = threadIdx.x & 31;          // wave32 lane
    const int wave = threadIdx.x >> 5;          // 8 waves per block
    const int b    = blockIdx.x * 8 + wave;     // batch element (uniform per wave)
    if (b >= batch) return;                     // wave-uniform -> EXEC all-ones inside

    // Lane t owns variable indices {t, t+32}; inequality rows {t, t+32} (lower,
    // d_i <= 1-x_i) and {t+64, t+96} (upper, -d_i <= x_i).
    const float c0 = c_in[b * N_VARS + lane];
    const float c1 = c_in[b * N_VARS + lane + 32];

    float x0 = 0.5f, x1 = 0.5f;
    float o0 = 0.0f, o1 = 0.0f;

    for (int it = 0; it <= SQP_IT; ++it) {
        // Linearized subproblem data at current x:
        //   p = x - c ; h = [1-x ; x] ; b_eq = 1 - sum(x)
        const float p0  = x0 - c0, p1 = x1 - c1;
        const float hL0 = 1.0f - x0, hL1 = 1.0f - x1;
        const float hU0 = x0,        hU1 = x1;
        const float beq = 1.0f - wave_sum(x0 + x1);

        // IPM state (all in registers): d, s, lam, nu
        float d0 = 0.0f, d1 = 0.0f;
        float sL0 = 1.0f, sL1 = 1.0f, sU0 = 1.0f, sU1 = 1.0f;
        float lL0 = 1.0f, lL1 = 1.0f, lU0 = 1.0f, lU1 = 1.0f;
        float nu  = 0.0f;

        for (int k = 0; k < IPM_IT; ++k) {
            // mu = sigma * sum(lam*s)/m ; r_pe = 1'd - b   (two independent WMMA sums)
            const float mu   = (SIGMA_ / (float)M_INEQ) *
                               wave_sum(lL0 * sL0 + lL1 * sL1 + lU0 * sU0 + lU1 * sU1);
            const float r_pe = wave_sum(d0 + d1) - beq;

            // reciprocals of the slacks (refined: they feed the solve)
            const float rsL0 = rcp_acc(sL0), rsL1 = rcp_acc(sL1);
            const float rsU0 = rcp_acc(sU0), rsU1 = rcp_acc(sU1);

            // r_d = Qd + p + G'lam + A'nu  (Q=I, G'=[I,-I], A'=1)
            const float rd0 = d0 + p0 + (lL0 - lU0) + nu;
            const float rd1 = d1 + p1 + (lL1 - lU1) + nu;

            // r_pi = Gd + s - h
            const float rpiL0 =  d0 + sL0 - hL0;
            const float rpiL1 =  d1 + sL1 - hL1;
            const float rpiU0 = -d0 + sU0 - hU0;
            const float rpiU1 = -d1 + sU1 - hU1;

            // r_c = lam*s - mu
            const float rcL0 = lL0 * sL0 - mu;
            const float rcL1 = lL1 * sL1 - mu;
            const float rcU0 = lU0 * sU0 - mu;
            const float rcU1 = lU1 * sU1 - mu;

            // w = (lam*r_pi - r_c)/s ; rhs_top = -(r_d + G'w)
            const float wL0 = (lL0 * rpiL0 - rcL0) * rsL0;
            const float wL1 = (lL1 * rpiL1 - rcL1) * rsL1;
            const float wU0 = (lU0 * rpiU0 - rcU0) * rsU0;
            const float wU1 = (lU1 * rpiU1 - rcU1) * rsU1;
            const float rt0 = -(rd0 + wL0 - wU0);
            const float rt1 = -(rd1 + wL1 - wU1);

            // M = I + G'diag(lam/s)G is DIAGONAL
            const float Md0 = 1.0f + lL0 * rsL0 + lU0 * rsU0;
            const float Md1 = 1.0f + lL1 * rsL1 + lU1 * rsU1;
            const float i0 = rcp_acc(Md0), i1 = rcp_acc(Md1);

            // Arrow-KKT closed form:  M dd + 1*dnu = rhs_top ; 1'dd = -r_pe
            const float num = wave_sum(rt0 * i0 + rt1 * i1);
            const float den = wave_sum(i0 + i1);
            const float dnu = (num + r_pe) * rcp_acc(den);
            const float dd0 = (rt0 - dnu) * i0;
            const float dd1 = (rt1 - dnu) * i1;

            // ds = -r_pi - G dd ; dlam = (-r_c - lam*ds)/s
            const float dsL0 = -rpiL0 - dd0;
            const float dsL1 = -rpiL1 - dd1;
            const float dsU0 = -rpiU0 + dd0;
            const float dsU1 = -rpiU1 + dd1;
            const float dlL0 = (-rcL0 - lL0 * dsL0) * rsL0;
            const float dlL1 = (-rcL1 - lL1 * dsL1) * rsL1;
            const float dlU0 = (-rcU0 - lU0 * dsU0) * rsU0;
            const float dlU1 = (-rcU1 - lU1 * dsU1) * rsU1;

            // fraction-to-boundary step: min over all 256 ratios (8 per lane)
            float r = fminf(fminf(ftb(sL0, dsL0), ftb(sL1, dsL1)),
                            fminf(ftb(sU0, dsU0), ftb(sU1, dsU1)));
            r = fminf(r, fminf(fminf(ftb(lL0, dlL0), ftb(lL1, dlL1)),
                               fminf(ftb(lU0, dlU0), ftb(lU1, dlU1))));
            r = wave_min(r);
            const float a = fminf(1.0f, 0.99f * r);

            d0 += a * dd0;  d1 += a * dd1;
            sL0 += a * dsL0; sL1 += a * dsL1; sU0 += a * dsU0; sU1 += a * dsU1;
            lL0 += a * dlL0; lL1 += a * dlL1; lU0 += a * dlU0; lU1 += a * dlU1;
            nu  += a * dnu;
        }

        if (it < SQP_IT) {              // SQP damped update
            x0 += ALPHA_ * d0;
            x1 += ALPHA_ * d1;
        } else {                        // final solve: out = x + d
            o0 = x0 + d0;
            o1 = x1 + d1;
        }
    }

    out[b * N_VARS + lane]      = o0;
    out[b * N_VARS + lane + 32] = o1;
}

extern "C" void kernel_launch(void* const* d_in, const int* in_sizes, int n_in,
                              void* d_out, int out_size, void* d_ws, size_t ws_size,
                              hipStream_t stream) {
    (void)n_in; (void)d_ws; (void)ws_size; (void)out_size;
    const float* c = (const float*)d_in[0];
    float* out = (float*)d_out;
    const int batch = in_sizes[0] / N_VARS;      // 2048
    const int blocks = (batch + 7) / 8;          // 8 waves (batch elems) / block
    FixedPtSQP_kernel<<<blocks, 256, 0, stream>>>(c, out, batch);
}